// MambaBlock_16003048145056
// MI455X (gfx1250) — compile-verified
//
#include <hip/hip_runtime.h>
#include <hip/hip_bf16.h>

// ---------------- problem constants (match reference) ----------------
#define BB      4
#define LL      2048
#define DMODEL  1024
#define DSTATE  16
#define DCONV   4
#define DINNER  2048          // EXPAND * DMODEL
#define DTRANK  64
#define XDBL_W  (DTRANK + 2*DSTATE)   // 96
#define NTOKENS ((size_t)BB * LL)     // 8192
#define EPSLN   1e-5f

typedef __attribute__((ext_vector_type(2))) float v2f;
typedef __attribute__((ext_vector_type(8))) float v8f;

__device__ __forceinline__ float fast_silu(float x) {
    return x / (1.f + __expf(-x));
}

// =====================================================================
// Kernel 1: LayerNorm, one block per token row (D = 1024)
// =====================================================================
__global__ void __launch_bounds__(256)
ln_kernel(const float* __restrict__ x, const float* __restrict__ w,
          const float* __restrict__ b, float* __restrict__ xn)
{
    const int row = blockIdx.x;
    const float* xr = x + (size_t)row * DMODEL;
    float* o = xn + (size_t)row * DMODEL;
    __shared__ float red[256];
    const int tid = threadIdx.x;

    float s = 0.f;
    for (int i = tid; i < DMODEL; i += 256) s += xr[i];
    red[tid] = s; __syncthreads();
    for (int st = 128; st > 0; st >>= 1) {
        if (tid < st) red[tid] += red[tid + st];
        __syncthreads();
    }
    const float mu = red[0] * (1.f / DMODEL);
    __syncthreads();

    float v = 0.f;
    for (int i = tid; i < DMODEL; i += 256) { float d = xr[i] - mu; v += d * d; }
    red[tid] = v; __syncthreads();
    for (int st = 128; st > 0; st >>= 1) {
        if (tid < st) red[tid] += red[tid + st];
        __syncthreads();
    }
    const float rstd = rsqrtf(red[0] * (1.f / DMODEL) + EPSLN);

    for (int i = tid; i < DMODEL; i += 256)
        o[i] = (xr[i] - mu) * rstd * w[i] + b[i];
}

// =====================================================================
// WMMA f32 GEMM:  C[M,N] = A[M,K(lda)] * W[N,K]^T   (+ epilogue)
//   EPI 0: plain store
//   EPI 1: v = softplus(v + aux[col])          (dt projection)
//   EPI 2: v = v + aux[row*ldc + col]          (residual add)
//
// One wave owns a 32 x (16*NT) output tile (MT=2 row tiles). K-loop
// steps by 8 with DISTANCE-2 register prefetch: fragments for k0+8 and
// k0+12 are issued before the two WMMA groups for k0 / k0+4, giving
// each load ~16 WMMAs of latency cover and keeping up to 12 loads in
// flight (s_wait_loadcnt floor stays high; XDL pipe never drains).
//
// Fragment maps (ISA 7.12.2, f32):
//  A 16x4 : lane<16 -> float2 A[m0+l16][k0..k0+1]; lane>=16 -> k0+2..k0+3
//  B 4x16 : lane holds W[n0+l16][k0 + 2*(lane>=16) ..] (W is [N,K] => B=W^T)
//  C/D    : VGPR r -> rows m0+r (lanes 0-15) / m0+r+8 (lanes 16-31)
// Requires K % 8 == 0 and K >= 8 (true for K = 1024, 2048, 64).
// =====================================================================
template<int NT, int EPI>
__global__ void __launch_bounds__(256)
gemm_wmma_f32(const float* __restrict__ A, int lda,
              const float* __restrict__ W,
              const float* __restrict__ aux,
              float* __restrict__ C, int ldc,
              int M, int N, int K)
{
    constexpr int MT = 2;
    const int lane = threadIdx.x & 31;
    const int wave = threadIdx.x >> 5;
    const int half = lane >> 4;        // 0 or 1
    const int l16  = lane & 15;

    const int tilesN = N / (16 * NT);
    const int totalTiles = (M / (16 * MT)) * tilesN;
    const int tile = blockIdx.x * (blockDim.x >> 5) + wave;
    if (tile >= totalTiles) return;     // wave-uniform: EXEC stays all-ones

    const int m0 = (tile / tilesN) * (16 * MT);
    const int n0 = (tile % tilesN) * (16 * NT);

    v8f acc[MT][NT] = {};
    const float* Ap[MT];
    const float* Wp[NT];
#pragma unroll
    for (int mt = 0; mt < MT; ++mt)
        Ap[mt] = A + (size_t)(m0 + mt * 16 + l16) * lda + 2 * half;
#pragma unroll
    for (int t = 0; t < NT; ++t)
        Wp[t] = W + (size_t)(n0 + t * 16 + l16) * K + 2 * half;

    // prologue: two live fragment sets (k = 0 and k = 4)
    v2f a0[MT], b0[NT], a1[MT], b1[NT];
#pragma unroll
    for (int mt = 0; mt < MT; ++mt) a0[mt] = *(const v2f*)(Ap[mt]);
#pragma unroll
    for (int t = 0; t < NT; ++t)    b0[t]  = *(const v2f*)(Wp[t]);
#pragma unroll
    for (int mt = 0; mt < MT; ++mt) a1[mt] = *(const v2f*)(Ap[mt] + 4);
#pragma unroll
    for (int t = 0; t < NT; ++t)    b1[t]  = *(const v2f*)(Wp[t] + 4);

    for (int k0 = 0; k0 < K; k0 += 8) {
        const int kp0 = (k0 + 8  < K) ? (k0 + 8)  : k0;  // clamp: tail reloads, unused
        const int kp1 = (k0 + 12 < K) ? (k0 + 12) : k0;

        // issue both prefetch groups up front (12 outstanding loads)
        v2f aF0[MT], bF0[NT], aF1[MT], bF1[NT];
#pragma unroll
        for (int mt = 0; mt < MT; ++mt) aF0[mt] = *(const v2f*)(Ap[mt] + kp0);
#pragma unroll
        for (int t = 0; t < NT; ++t)    bF0[t]  = *(const v2f*)(Wp[t] + kp0);
#pragma unroll
        for (int mt = 0; mt < MT; ++mt) aF1[mt] = *(const v2f*)(Ap[mt] + kp1);
#pragma unroll
        for (int t = 0; t < NT; ++t)    bF1[t]  = *(const v2f*)(Wp[t] + kp1);

        // compute group for k0
#pragma unroll
        for (int mt = 0; mt < MT; ++mt)
#pragma unroll
            for (int t = 0; t < NT; ++t)
                acc[mt][t] = __builtin_amdgcn_wmma_f32_16x16x4_f32(
                    false, a0[mt], false, b0[t], (short)0, acc[mt][t], false, false);

        // compute group for k0 + 4
#pragma unroll
        for (int mt = 0; mt < MT; ++mt)
#pragma unroll
            for (int t = 0; t < NT; ++t)
                acc[mt][t] = __builtin_amdgcn_wmma_f32_16x16x4_f32(
                    false, a1[mt], false, b1[t], (short)0, acc[mt][t], false, false);

        // rotate buffers (renamed away by the register allocator)
#pragma unroll
        for (int mt = 0; mt < MT; ++mt) { a0[mt] = aF0[mt]; a1[mt] = aF1[mt]; }
#pragma unroll
        for (int t = 0; t < NT; ++t)    { b0[t]  = bF0[t];  b1[t]  = bF1[t];  }
    }

#pragma unroll
    for (int mt = 0; mt < MT; ++mt) {
#pragma unroll
        for (int t = 0; t < NT; ++t) {
            const int col = n0 + t * 16 + l16;
            const float bv = (EPI == 1) ? aux[col] : 0.f;
#pragma unroll
            for (int r = 0; r < 8; ++r) {
                const int row = m0 + mt * 16 + r + 8 * half;
                float v = acc[mt][t][r];
                if (EPI == 1) {
                    v += bv;
                    v = (v > 20.f) ? v : log1pf(__expf(v));   // softplus
                }
                if (EPI == 2) {
                    v += aux[(size_t)row * ldc + col];
                }
                C[(size_t)row * ldc + col] = v;
            }
        }
    }
}

// =====================================================================
// Kernel 3: depthwise causal conv1d (taps=4) + bias + SiLU
// x_in = xz[:, 0:DINNER] (row stride 2*DINNER) -> xc [NTOKENS, DINNER]
// (pure-bandwidth kernel; lanes coalesce along d)
// =====================================================================
__global__ void __launch_bounds__(256)
conv_silu_kernel(const float* __restrict__ xz, const float* __restrict__ cw,
                 const float* __restrict__ cb, float* __restrict__ xc)
{
    const size_t idx = (size_t)blockIdx.x * 256 + threadIdx.x;
    if (idx >= NTOKENS * (size_t)DINNER) return;
    const int d  = (int)(idx % DINNER);
    const size_t bl = idx / DINNER;
    const int l  = (int)(bl % LL);
    const size_t bbase = (bl - l) * (size_t)(2 * DINNER);   // (b*L)*4096

    float acc = cb[d];
    const float* wrow = cw + (size_t)d * DCONV;
#pragma unroll
    for (int j = 0; j < DCONV; ++j) {
        const int ls = l - (DCONV - 1) + j;
        if (ls >= 0)
            acc += wrow[j] * xz[bbase + (size_t)ls * (2 * DINNER) + d];
    }
    xc[bl * DINNER + d] = fast_silu(acc);
}

// =====================================================================
// Kernel 6: selective scan. Thread = one (b, d) channel, 16 states in
// registers. B_t/C_t handled wave-locally: lane i (0-15) loads B[i],
// lane 16+i loads C[i]; __shfl broadcasts inside the state loop — no
// block barriers, no LDS. Next timestep's operands are prefetched so
// the 16-deep exp/fma chain overlaps the loads.
// Fused D-term + SiLU gate; gated y written into the (consumed) x_in
// half of xz.  grid = BB * (DINNER/256)
// =====================================================================
__global__ void __launch_bounds__(256)
scan_kernel(const float* __restrict__ xdbl, const float* __restrict__ dtb,
            const float* __restrict__ xc,   const float* __restrict__ Alog,
            const float* __restrict__ Dp,   const float* __restrict__ xz_z,
            float* __restrict__ yg /* xz base, row stride 2*DINNER */)
{
    const int tid  = threadIdx.x;
    const int lane = tid & 31;
    const int chunks = DINNER / 256;
    const int b = blockIdx.x / chunks;
    const int d = (blockIdx.x % chunks) * 256 + tid;

    float Ad[DSTATE];
#pragma unroll
    for (int s = 0; s < DSTATE; ++s)
        Ad[s] = -__expf(Alog[(size_t)d * DSTATE + s]);

    float h[DSTATE];
#pragma unroll
    for (int s = 0; s < DSTATE; ++s) h[s] = 0.f;

    const float dval = Dp[d];
    const size_t base = (size_t)b * LL;

    // prologue loads for l = 0
    float bc  = xdbl[base * XDBL_W + DTRANK + lane];
    float dtv = dtb [base * DINNER + d];
    float u   = xc  [base * DINNER + d];
    float z   = xz_z[base * (2 * DINNER) + DINNER + d];

    for (int l = 0; l < LL; ++l) {
        // prefetch next timestep (clamped on the last iteration)
        const size_t bln = base + ((l + 1 < LL) ? (l + 1) : l);
        const float bc_n  = xdbl[bln * XDBL_W + DTRANK + lane];
        const float dt_n  = dtb [bln * DINNER + d];
        const float u_n   = xc  [bln * DINNER + d];
        const float z_n   = xz_z[bln * (2 * DINNER) + DINNER + d];

        const float du = dtv * u;
        float y = 0.f;
#pragma unroll
        for (int s = 0; s < DSTATE; ++s) {
            const float Bs = __shfl(bc, s, 32);
            const float Cs = __shfl(bc, s + DSTATE, 32);
            const float dA = __expf(dtv * Ad[s]);
            h[s] = fmaf(dA, h[s], du * Bs);
            y    = fmaf(h[s], Cs, y);
        }
        const size_t bl = base + l;
        yg[bl * (2 * DINNER) + d] = (y + u * dval) * fast_silu(z);

        bc = bc_n; dtv = dt_n; u = u_n; z = z_n;
    }
}

// =====================================================================
// Host-side launcher
// =====================================================================
extern "C" void kernel_launch(void* const* d_in, const int* in_sizes, int n_in,
                              void* d_out, int out_size, void* d_ws, size_t ws_size,
                              hipStream_t stream)
{
    const float* x      = (const float*)d_in[0];
    const float* ln_w   = (const float*)d_in[1];
    const float* ln_b   = (const float*)d_in[2];
    const float* W_in   = (const float*)d_in[3];   // [4096, 1024]
    const float* conv_w = (const float*)d_in[4];   // [2048, 1, 4]
    const float* conv_b = (const float*)d_in[5];
    const float* W_x    = (const float*)d_in[6];   // [96, 2048]
    const float* W_dt   = (const float*)d_in[7];   // [2048, 64]
    const float* b_dt   = (const float*)d_in[8];
    const float* A_log  = (const float*)d_in[9];   // [2048, 16]
    const float* Dp     = (const float*)d_in[10];
    const float* W_out  = (const float*)d_in[11];  // [1024, 2048]
    float* out = (float*)d_out;

    // workspace layout (floats)
    float* ws = (float*)d_ws;
    float* xn   = ws;                                  // 8192*1024
    float* xz   = xn   + NTOKENS * DMODEL;             // 8192*4096
    float* xc   = xz   + NTOKENS * 2 * DINNER;         // 8192*2048
    float* xdbl = xc   + NTOKENS * DINNER;             // 8192*96
    float* dtb  = xdbl + NTOKENS * XDBL_W;             // 8192*2048

    // 1) LayerNorm
    ln_kernel<<<(int)NTOKENS, 256, 0, stream>>>(x, ln_w, ln_b, xn);

    // 2) in_proj: xz = xn * W_in^T   [8192 x 4096], K=1024  (MT=2, NT=4)
    {
        const int tiles = (8192 / 32) * (4096 / 64);
        gemm_wmma_f32<4, 0><<<tiles / 8, 256, 0, stream>>>(
            xn, DMODEL, W_in, nullptr, xz, 2 * DINNER, 8192, 2 * DINNER, DMODEL);
    }

    // 3) causal depthwise conv + SiLU -> xc
    {
        const size_t n = NTOKENS * (size_t)DINNER;
        conv_silu_kernel<<<(int)((n + 255) / 256), 256, 0, stream>>>(xz, conv_w, conv_b, xc);
    }

    // 4) x_dbl = xc * W_x^T   [8192 x 96], K=2048  (MT=2, NT=2 -> 96 = 3*32)
    {
        const int tiles = (8192 / 32) * (XDBL_W / 32);
        gemm_wmma_f32<2, 0><<<tiles / 8, 256, 0, stream>>>(
            xc, DINNER, W_x, nullptr, xdbl, XDBL_W, 8192, XDBL_W, DINNER);
    }

    // 5) dt = softplus(dt_r * W_dt^T + b_dt)   [8192 x 2048], K=64, lda=96
    {
        const int tiles = (8192 / 32) * (DINNER / 64);
        gemm_wmma_f32<4, 1><<<tiles / 8, 256, 0, stream>>>(
            xdbl, XDBL_W, W_dt, b_dt, dtb, DINNER, 8192, DINNER, DTRANK);
    }

    // 6) selective scan + D-term + SiLU gate -> gated y into xz[:, 0:2048]
    scan_kernel<<<BB * (DINNER / 256), 256, 0, stream>>>(
        xdbl, dtb, xc, A_log, Dp, xz, xz);

    // 7) out = yg * W_out^T + x   [8192 x 1024], K=2048, lda=4096
    {
        const int tiles = (8192 / 32) * (DMODEL / 64);
        gemm_wmma_f32<4, 2><<<tiles / 8, 256, 0, stream>>>(
            xz, 2 * DINNER, W_out, x, out, DMODEL, 8192, DMODEL, DINNER);
    }
}